// ModulatedDeformConv_26551487824158
// MI455X (gfx1250) — compile-verified
//
#include <hip/hip_runtime.h>
#include <hip/hip_bf16.h>
#include <math.h>

// ---------------------------------------------------------------------------
// Modulated deformable conv (DCNv2) for MI455X (gfx1250, wave32).
//  Stage 1: 3x3 conv y -> offmap(27ch).  Raw y tile [192x64] pulled into LDS
//           by the Tensor Data Mover (tensor_load_to_lds, 3D tile descriptor);
//           3x3 kj realized as column shifts; fp32 WMMA GEMM (Mpad=32, K=576).
//  Stage 2: bilinear sampling -> LDS [576 x 64px], fp32 WMMA GEMM (M=64,K=576)
//           with 32-row M-supertiles (2 accumulators / wave).
// fp32 V_WMMA_F32_16X16X4_F32 keeps reference precision; problem is compute
// bound (~15 GFLOP vs ~130 MB -> 5.6us HBM floor at 23.3 TB/s).
// ---------------------------------------------------------------------------

typedef float v2f  __attribute__((ext_vector_type(2)));
typedef float v8f  __attribute__((ext_vector_type(8)));
typedef unsigned int u32x4 __attribute__((ext_vector_type(4)));
typedef int   i32x8 __attribute__((ext_vector_type(8)));
typedef int   i32x4 __attribute__((ext_vector_type(4)));

static constexpr int BN   = 2;
static constexpr int CI   = 64;
static constexpr int CO   = 64;
static constexpr int IH   = 256;
static constexpr int IW   = 256;
static constexpr int KK   = 9;          // 3*3
static constexpr int KDIM = CI * KK;    // 576
static constexpr int OCH  = 3 * KK;     // 27
static constexpr int HW   = IH * IW;
static constexpr int TR   = CI * 3;     // 192 raw tile rows (c,ki)

__device__ __forceinline__ v8f wmma_f32(v2f a, v2f b, v8f c) {
  return __builtin_amdgcn_wmma_f32_16x16x4_f32(
      false, a, false, b, (short)0, c, false, false);
}

__device__ __forceinline__ void wait_tensorcnt0() {
#if defined(__gfx1250__)
#if __has_builtin(__builtin_amdgcn_s_wait_tensorcnt)
  __builtin_amdgcn_s_wait_tensorcnt(0);
#else
  asm volatile("s_wait_tensorcnt 0x0" ::: "memory");
#endif
#endif
}

#if defined(__gfx1250__) && __has_builtin(__builtin_amdgcn_tensor_load_to_lds)
#define DCN_HAS_TDM 1
#else
#define DCN_HAS_TDM 0
#endif

// ===========================================================================
// Stage 1: offset/mask conv.  Block = (b, row i, 64-px strip).
// LDS: tile[192][64] raw y  (+ left/right halo columns, 192 each).
// ===========================================================================
static constexpr int NP1      = 64;
static constexpr int LCOL_OFF = TR * NP1;        // 12288 floats
static constexpr int RCOL_OFF = LCOL_OFF + TR;   // 12480
static constexpr int LDS1_FLT = RCOL_OFF + TR;   // 12672 -> 50688 B

__global__ __launch_bounds__(256)
void dcn_offset_conv_kernel(const float* __restrict__ y,
                            const float* __restrict__ w_off,
                            const float* __restrict__ b_off,
                            float* __restrict__ offmap) {
  extern __shared__ float Ls[];
  const int b  = blockIdx.z;
  const int i  = blockIdx.y;
  const int j0 = blockIdx.x * NP1;
  const int tid  = threadIdx.x;
  const int wave = tid >> 5;

  const bool interior = (i > 0) && (i < IH - 1);
  bool tdm_used = false;

#if DCN_HAS_TDM
  if (interior) {
    tdm_used = true;
    if (wave == 0) {
      // --- Tensor DMA descriptor (ISA 8.3/8.4): 3D tile ---
      //   dim0 = W (x, 64 cols @ stride 1)
      //   dim1 = H rows (3 rows @ stride 256)
      //   dim2 = channels (64 @ stride 65536)   -> LDS [(c*3+ki)*64 + x]
      unsigned long long ga =
          (unsigned long long)(const void*)(y + ((size_t)b * CI * IH + (i - 1)) * IW + j0);
      u32x4 g0;
      g0.x = 1u;                                   // count=1, load, user mode
      g0.y = 0u;                                   // lds_addr = 0 (tile base)
      g0.z = (unsigned)(ga & 0xFFFFFFFFu);         // global_addr[31:0]
      g0.w = (unsigned)((ga >> 32) & 0x1FFFFFFu)   // global_addr[56:32]
             | (2u << 30);                         // type = 2 ("image")
      i32x8 g1;
      g1[0] = (2 << 16);                           // data_size = 4B
      g1[1] = (IW & 0xFFFF) << 16;                 // tensor_dim0 lo16 (256)
      g1[2] = ((IW >> 16) & 0xFFFF) | ((IH & 0xFFFF) << 16);  // d0 hi | d1 lo
      g1[3] = ((IH >> 16) & 0xFFFF) | (NP1 << 16); // d1 hi | tile_dim0 = 64
      g1[4] = 3 | (CI << 16);                      // tile_dim1=3 | tile_dim2=64
      g1[5] = IW;                                  // tensor_dim0_stride lo32
      g1[6] = 0 | ((HW & 0xFFFF) << 16);           // d0s hi16 | d1 stride lo16
      g1[7] = (HW >> 16);                          // d1 stride [47:16]
      i32x4 g2; g2[0] = BN * CI; g2[1] = 0; g2[2] = 0; g2[3] = 0;  // dim2 len
      i32x4 g3; g3[0] = 0; g3[1] = 0; g3[2] = 0; g3[3] = 0;
      i32x8 g4; g4[0] = 0; g4[1] = 0; g4[2] = 0; g4[3] = 0;
                g4[4] = 0; g4[5] = 0; g4[6] = 0; g4[7] = 0;
      // 6-arg toolchain form: (g0, g1, g2, g3, extra_group, cpol)
      __builtin_amdgcn_tensor_load_to_lds(g0, g1, g2, g3, g4, 0);
      wait_tensorcnt0();
    }
  }
#endif

  // ---- halo columns (j0-1 and j0+64) + vertical zero handling ----
  for (int t = tid; t < TR; t += blockDim.x) {
    const int c = t / 3, ki = t - 3 * c;
    const int row = i - 1 + ki;
    const bool rok = (row >= 0) && (row < IH);
    const float* src = y + ((size_t)b * CI + c) * HW + (size_t)(rok ? row : 0) * IW;
    Ls[LCOL_OFF + t] = (rok && (j0 - 1 >= 0))  ? src[j0 - 1]  : 0.0f;
    Ls[RCOL_OFF + t] = (rok && (j0 + NP1 < IW)) ? src[j0 + NP1] : 0.0f;
  }

  // ---- fallback main-tile fill (edge rows, or no TDM builtin) ----
  if (!tdm_used) {
    for (int e = tid; e < TR * NP1; e += blockDim.x) {
      const int t = e >> 6, col = e & (NP1 - 1);
      const int c = t / 3, ki = t - 3 * c;
      const int row = i - 1 + ki;
      const bool ok = (row >= 0) && (row < IH);
      Ls[t * NP1 + col] =
          ok ? y[((size_t)b * CI + c) * HW + (size_t)row * IW + j0 + col] : 0.0f;
    }
  }
  __syncthreads();

  // ---- WMMA GEMM: out[32pad x 64px]; K split as (t=192) x (kj=3) ----
  const int lane  = tid & 31;
  const int m0 = (wave & 1) * 16;                 // oc tile
  const int n0 = (wave >> 1) * 16;                // pixel tile
  const int half  = lane >> 4;
  const int half2 = half * 2;
  const int mlane = m0 + (lane & 15);
  const int mc = mlane > (OCH - 1) ? (OCH - 1) : mlane;
  const int n  = n0 + (lane & 15);

  v8f acc = {};
  for (int tt = 0; tt < TR; tt += 4) {
    #pragma unroll
    for (int kj = 0; kj < 3; ++kj) {
      const int cj = n + kj - 1;                  // column in [-1, 64]
      // LDS source: halo arrays for cj==-1 / cj==64, else main tile
      const float* bp0;
      if (cj < 0)          bp0 = &Ls[LCOL_OFF + tt + half2];
      else if (cj >= NP1)  bp0 = &Ls[RCOL_OFF + tt + half2];
      else                 bp0 = &Ls[(tt + half2) * NP1 + cj];
      const int bstride = (cj < 0 || cj >= NP1) ? 1 : NP1;
      v2f bf; bf.x = bp0[0]; bf.y = bp0[bstride];
      // A frag: w_off[oc][(t)*3 + kj], t = tt + r + 2*half  (stride 12B)
      const float* ap = w_off + (size_t)mc * KDIM + (tt + half2) * 3 + kj;
      v2f a; a.x = ap[0]; a.y = ap[3];
      acc = wmma_f32(a, bf, acc);
    }
  }

  const int j = j0 + n;
  #pragma unroll
  for (int r = 0; r < 8; ++r) {
    const int oc = m0 + r + 8 * half;
    if (oc < OCH)
      offmap[((size_t)b * OCH + oc) * HW + (size_t)i * IW + j] = acc[r] + b_off[oc];
  }
}

// ===========================================================================
// Stage 2: deformable conv.  Block = (b, row i, 64-px strip), 64 oc.
// LDS: sampled B matrix [576][65].  8 waves = 2 M-supertiles(32) x 4 n-tiles.
// ===========================================================================
static constexpr int NP2   = 64;
static constexpr int BSTR2 = 65;

__global__ __launch_bounds__(256)
void dcn_main_kernel(const float* __restrict__ x,
                     const float* __restrict__ offmap,
                     const float* __restrict__ w_dcn,
                     const float* __restrict__ b_dcn,
                     float* __restrict__ out) {
  extern __shared__ float Bs[];                   // [KDIM][65]
  const int b  = blockIdx.z;
  const int i  = blockIdx.y;
  const int j0 = blockIdx.x * NP2;
  const int tid = threadIdx.x;

  // ---- sampling: 576 (k,pixel) pairs; coords once, 64-channel gather ----
  for (int pr = tid; pr < KK * NP2; pr += blockDim.x) {
    const int pix = pr & (NP2 - 1);
    const int k   = pr >> 6;
    const int j   = j0 + pix;
    const size_t obase = ((size_t)b * OCH * IH + i) * IW + j;
    // reference mapping: dy_k = ch 2k, dx_k = ch 2k+1, mask_k = ch 18+k
    const float dy = offmap[obase + (size_t)(2 * k    ) * HW];
    const float dx = offmap[obase + (size_t)(2 * k + 1) * HW];
    const float mr = offmap[obase + (size_t)(18 + k   ) * HW];
    const float mask = 1.0f / (1.0f + __expf(-mr));

    const int ki = k / 3, kj = k - 3 * ki;
    const float py = (float)(i - 1 + ki) + dy;
    const float px = (float)(j - 1 + kj) + dx;
    const float y0f = floorf(py), x0f = floorf(px);
    const float wy = py - y0f,   wx = px - x0f;
    const int iy0 = (int)y0f, ix0 = (int)x0f;
    const int iy1 = iy0 + 1,  ix1 = ix0 + 1;
    const float vy0 = (iy0 >= 0 && iy0 < IH) ? 1.0f : 0.0f;
    const float vy1 = (iy1 >= 0 && iy1 < IH) ? 1.0f : 0.0f;
    const float vx0 = (ix0 >= 0 && ix0 < IW) ? 1.0f : 0.0f;
    const float vx1 = (ix1 >= 0 && ix1 < IW) ? 1.0f : 0.0f;
    const int y0c = min(max(iy0, 0), IH - 1), y1c = min(max(iy1, 0), IH - 1);
    const int x0c = min(max(ix0, 0), IW - 1), x1c = min(max(ix1, 0), IW - 1);
    const float w00 = (1.0f - wy) * (1.0f - wx) * vy0 * vx0 * mask;
    const float w01 = (1.0f - wy) * wx          * vy0 * vx1 * mask;
    const float w10 = wy          * (1.0f - wx) * vy1 * vx0 * mask;
    const float w11 = wy          * wx          * vy1 * vx1 * mask;
    const int o00 = y0c * IW + x0c, o01 = y0c * IW + x1c;
    const int o10 = y1c * IW + x0c, o11 = y1c * IW + x1c;

    const float* xb = x + (size_t)b * CI * HW;
    #pragma unroll 4
    for (int c = 0; c < CI; ++c) {
      const float* xc = xb + (size_t)c * HW;
      Bs[(c * KK + k) * BSTR2 + pix] =
          w00 * xc[o00] + w01 * xc[o01] + w10 * xc[o10] + w11 * xc[o11];
    }
  }
  __syncthreads();

  // ---- WMMA GEMM: out[64oc x 64px] = Wdcn[64x576] * Bs[576x64] ----
  // Wave job: 32-row M-supertile (2 accumulators, shared B frag) x 16-px tile
  const int wave = tid >> 5, lane = tid & 31;
  const int ms = (wave & 1) * 32;                 // M-supertile
  const int n0 = (wave >> 1) * 16;                // pixel tile
  const int half  = lane >> 4;
  const int half2 = half * 2;
  const int mA = ms + (lane & 15);
  const int mB = mA + 16;
  const int n  = n0 + (lane & 15);
  const float2* W2 = reinterpret_cast<const float2*>(w_dcn);

  v8f acc0 = {}, acc1 = {};
  #pragma unroll 4
  for (int kk = 0; kk < KDIM; kk += 4) {
    const float2 afA = W2[(mA * KDIM + kk + half2) >> 1];
    const float2 afB = W2[(mB * KDIM + kk + half2) >> 1];
    v2f a0; a0.x = afA.x; a0.y = afA.y;
    v2f a1; a1.x = afB.x; a1.y = afB.y;
    v2f bf;
    bf.x = Bs[(kk + half2    ) * BSTR2 + n];
    bf.y = Bs[(kk + half2 + 1) * BSTR2 + n];
    acc0 = wmma_f32(a0, bf, acc0);                // two independent chains
    acc1 = wmma_f32(a1, bf, acc1);
  }

  const int j = j0 + n;
  #pragma unroll
  for (int r = 0; r < 8; ++r) {
    const int ocA = ms + r + 8 * half;
    const int ocB = ocA + 16;
    out[((size_t)b * CO + ocA) * HW + (size_t)i * IW + j] = acc0[r] + b_dcn[ocA];
    out[((size_t)b * CO + ocB) * HW + (size_t)i * IW + j] = acc1[r] + b_dcn[ocB];
  }
}

// ===========================================================================
extern "C" void kernel_launch(void* const* d_in, const int* in_sizes, int n_in,
                              void* d_out, int out_size, void* d_ws, size_t ws_size,
                              hipStream_t stream) {
  const float* x     = (const float*)d_in[0];
  const float* y     = (const float*)d_in[1];
  const float* w_off = (const float*)d_in[2];
  const float* b_off = (const float*)d_in[3];
  const float* w_dcn = (const float*)d_in[4];
  const float* b_dcn = (const float*)d_in[5];
  float* out    = (float*)d_out;
  float* offmap = (float*)d_ws;   // 2*27*256*256 f32 = 13.5 MB intermediate

  // Stage 1: TDM-fed raw tile.  LDS = 12672 floats = 50,688 B.
  dim3 g1(IW / NP1, IH, BN);
  dcn_offset_conv_kernel<<<g1, 256, LDS1_FLT * sizeof(float), stream>>>(
      y, w_off, b_off, offmap);

  // Stage 2: sampled GEMM.  LDS = 576*65*4 = 149,760 B (WGP has 320 KB).
  dim3 g2(IW / NP2, IH, BN);
  dcn_main_kernel<<<g2, 256, KDIM * BSTR2 * sizeof(float), stream>>>(
      x, offmap, w_dcn, b_dcn, out);
}